// VectorsFromMaskV2_83760452206671
// MI455X (gfx1250) — compile-verified
//
#include <hip/hip_runtime.h>
#include <hip/hip_bf16.h>
#include <stdint.h>

// Problem constants (from reference): B=8, C=128, H*W=65536, V=32
#define B_      8
#define C_      128
#define P_      65536
#define V_      32

#define CH      8                    // channels per block
#define PT      8192                 // pixels per block tile
#define SPLITS  (P_ / PT)            // 8 pixel tiles per image
#define NTHR    256                  // 8 waves (wave32)
#define NWAVES  (NTHR / 32)
#define ITERS   (PT / (NTHR * 4))    // 8 float4-iterations per thread
#define TSTRIDE 33                   // odd row stride -> seg ids map to distinct LDS banks

__global__ __launch_bounds__(NTHR)
void seg_out_init(float* __restrict__ out, int n) {
  int i = blockIdx.x * NTHR + threadIdx.x;
  if (i < n) out[i] = -__builtin_inff();
}

__global__ __launch_bounds__(NTHR)
void seg_max_kernel(const float* __restrict__ enc,
                    const int*   __restrict__ masks,
                    float*       __restrict__ out) {
  __shared__ float s_tab[NWAVES * CH * TSTRIDE];   // 8448 B: per-wave max tables
  __shared__ int   s_mask[PT];                     // 32 KB: async-staged mask tile

  const int tid  = threadIdx.x;
  const int wave = tid >> 5;                       // wave32
  const int tile = blockIdx.x;                     // pixel tile
  const int cg   = blockIdx.y;                     // channel group
  const int b    = blockIdx.z;                     // batch
  const int pixBase = tile * PT;
  const int cBase   = cg * CH;

  // ---- init per-wave seg-max tables to -inf ----
  #pragma unroll
  for (int i = tid; i < NWAVES * CH * TSTRIDE; i += NTHR)
    s_tab[i] = -__builtin_inff();

  // ---- async-stage the 8192-pixel mask tile into LDS (ASYNCcnt path) ----
  const unsigned smaskAddr = (unsigned)(uintptr_t)&s_mask[0];
  const int* mbase = masks + (size_t)b * P_ + pixBase;
  #pragma unroll
  for (int j = 0; j < ITERS; ++j) {                // 8 x b128 per thread
    int idx4 = tid + j * NTHR;
    unsigned ldsDst = smaskAddr + (unsigned)idx4 * 16u;
    unsigned long long ga = (unsigned long long)(uintptr_t)(mbase + idx4 * 4);
    asm volatile("global_load_async_to_lds_b128 %0, %1, off"
                 :: "v"(ldsDst), "v"(ga) : "memory");
  }
  asm volatile("s_wait_asynccnt 0" ::: "memory");
  __syncthreads();

  // ---- read mask ONCE into registers as pre-shifted table byte-offsets ----
  const int4* smask4 = (const int4*)s_mask;
  int4 moff[ITERS];
  #pragma unroll
  for (int it = 0; it < ITERS; ++it) {
    int4 mm = smask4[tid + it * NTHR];             // single ds_load_b128 pass
    moff[it].x = mm.x << 2;
    moff[it].y = mm.y << 2;
    moff[it].z = mm.z << 2;
    moff[it].w = mm.w << 2;
  }

  // ---- stream encoded, scatter-max into this wave's LDS table ----
  // NOTE: no "memory" clobber on ds_max -> compiler can cluster the 8
  // global_load_b128 per channel ahead of the scatter (deep MLP).
  const unsigned tabAddr = (unsigned)(uintptr_t)&s_tab[0];
  const float* encBase = enc + (size_t)(b * C_ + cBase) * P_ + pixBase;

  for (int c = 0; c < CH; ++c) {
    const float4* ep = (const float4*)(encBase + (size_t)c * P_);
    const unsigned rowAddr =
        tabAddr + (unsigned)((wave * CH + c) * TSTRIDE) * 4u;
    #pragma unroll
    for (int it = 0; it < ITERS; ++it) {
      int idx4 = tid + it * NTHR;
      float4 vv = ep[idx4];                        // coalesced b128 stream
      // speculative prefetch of next channel's line (OOB at the very end is
      // silently dropped by hardware -> branch-free)
      __builtin_prefetch((const void*)(ep + (size_t)(P_ / 4) + idx4), 0, 0);
      unsigned a0 = rowAddr + (unsigned)moff[it].x;
      unsigned a1 = rowAddr + (unsigned)moff[it].y;
      unsigned a2 = rowAddr + (unsigned)moff[it].z;
      unsigned a3 = rowAddr + (unsigned)moff[it].w;
      asm volatile("ds_max_num_f32 %0, %1" :: "v"(a0), "v"(vv.x));
      asm volatile("ds_max_num_f32 %0, %1" :: "v"(a1), "v"(vv.y));
      asm volatile("ds_max_num_f32 %0, %1" :: "v"(a2), "v"(vv.z));
      asm volatile("ds_max_num_f32 %0, %1" :: "v"(a3), "v"(vv.w));
    }
  }

  // our asm DS ops are invisible to the compiler's waitcnt insertion:
  asm volatile("s_wait_dscnt 0" ::: "memory");
  __syncthreads();

  // ---- merge 8 wave tables & publish: 256 threads == CH(8) x V(32) ----
  {
    int c  = tid >> 5;            // 0..7
    int v  = (tid & 31) + 1;      // seg ids 1..32
    float r = -__builtin_inff();
    #pragma unroll
    for (int w = 0; w < NWAVES; ++w)
      r = fmaxf(r, s_tab[(w * CH + c) * TSTRIDE + v]);
    float* op = out + ((size_t)(b * C_ + cBase + c) * V_ + (v - 1));
    asm volatile("global_atomic_max_num_f32 %0, %1, off"
                 :: "v"((unsigned long long)(uintptr_t)op), "v"(r) : "memory");
  }
}

extern "C" void kernel_launch(void* const* d_in, const int* in_sizes, int n_in,
                              void* d_out, int out_size, void* d_ws, size_t ws_size,
                              hipStream_t stream) {
  const float* enc   = (const float*)d_in[0];   // (8,128,256,256) fp32
  const int*   masks = (const int*)d_in[1];     // (8,1,256,256) int32 in [1,32]
  // d_in[2] = num_segments scalar (compile-time constant V_=32 here)
  float* out = (float*)d_out;                   // (8,128,32,1) fp32 = 32768

  seg_out_init<<<(out_size + NTHR - 1) / NTHR, NTHR, 0, stream>>>(out, out_size);

  dim3 grid(SPLITS, C_ / CH, B_);               // 8 x 16 x 8 = 1024 blocks
  seg_max_kernel<<<grid, NTHR, 0, stream>>>(enc, masks, out);
}